// DGP_RF_Embeddings_23862838297354
// MI455X (gfx1250) — compile-verified
//
#include <hip/hip_runtime.h>

// Problem constants (from the reference).
#define N_ROWS 500000
#define D_IN   64
#define NUM_RF 128
#define D_OUT  64
#define U_SEG  50000
#define EPSF   1e-8f
// scale = sqrt(2/NUM_RF) = sqrt(1/64) = 1/8 exactly.
#define RF_SCALE  0.125f
#define RF_SCALE2 0.015625f

// Fragment counts: layer1 B frags = 8 ntiles * 16 ksteps * 32 lanes = 4096
//                  layer2 B frags = 4 ntiles * 32 ksteps * 32 lanes = 4096
#define FRAG1_PAIRS 4096
#define FRAG2_PAIRS 4096

typedef __attribute__((ext_vector_type(2))) float v2f;
typedef __attribute__((ext_vector_type(8))) float v8f;

// D = A(16x4 f32) * B(4x16 f32) + C(16x16 f32), wave32, one matrix per wave.
__device__ __forceinline__ v8f wmma_f32_16x16x4(v2f a, v2f b, v8f c) {
    return __builtin_amdgcn_wmma_f32_16x16x4_f32(
        /*neg_a=*/false, a, /*neg_b=*/false, b,
        /*c_mod=*/(short)0, c, /*reuse_a=*/false, /*reuse_b=*/false);
}

// Pre-swizzle the weight matrices into per-lane WMMA B-fragment order so the
// hot loop does one coalesced global_load_b64 per fragment instead of two
// strided global_load_b32. Layout: [ntile][kstep][lane] -> (W[k,n], W[k+1,n])
// with k = 4*kstep + 2*(lane>=16), n = 16*ntile + lane%16.
__global__ __launch_bounds__(256) void dgp_fragmentize_weights(
    const float* __restrict__ Wmu1, const float* __restrict__ Wvar1,
    const float* __restrict__ Wmu2, const float* __restrict__ Wvar2,
    v2f* __restrict__ fMu1, v2f* __restrict__ fVar1,
    v2f* __restrict__ fMu2, v2f* __restrict__ fVar2)
{
    const int t = blockIdx.x * 256 + threadIdx.x;   // 0 .. 8191
    if (t < FRAG1_PAIRS) {
        const int f    = t;
        const int lane = f & 31;
        const int kk   = (f >> 5) & 15;
        const int n1   = f >> 9;
        const int k    = kk * 4 + (lane >> 4) * 2;
        const int n    = n1 * 16 + (lane & 15);
        v2f a, b;
        a.x = Wmu1[k * NUM_RF + n];
        a.y = Wmu1[(k + 1) * NUM_RF + n];
        b.x = Wvar1[k * NUM_RF + n];
        b.y = Wvar1[(k + 1) * NUM_RF + n];
        fMu1[f]  = a;
        fVar1[f] = b;
    } else {
        const int g    = t - FRAG1_PAIRS;
        const int lane = g & 31;
        const int kk   = (g >> 5) & 31;
        const int n2   = g >> 10;
        const int k    = kk * 4 + (lane >> 4) * 2;
        const int n    = n2 * 16 + (lane & 15);
        v2f a, b;
        a.x = Wmu2[k * D_OUT + n];
        a.y = Wmu2[(k + 1) * D_OUT + n];
        b.x = Wvar2[k * D_OUT + n];
        b.y = Wvar2[(k + 1) * D_OUT + n];
        fMu2[g]  = a;
        fVar2[g] = b;
    }
}

// Fused: RF layer (WMMA) -> ReLU gate -> output layer (WMMA) -> precision-
// weighted atomic segment accumulation. 2 waves per block, 16 rows per wave.
// FRAG=true reads pre-swizzled B fragments (1 coalesced b64/frag);
// FRAG=false reads the raw [K,N] weights directly (fallback, no d_ws needed).
template <bool FRAG>
__global__ __launch_bounds__(64) void dgp_fused_kernel(
    const float* __restrict__ X, const int* __restrict__ Xidx,
    const float* __restrict__ Wmu1, const float* __restrict__ Wvar1,
    const float* __restrict__ Wmu2, const float* __restrict__ Wvar2,
    const v2f* __restrict__ fMu1, const v2f* __restrict__ fVar1,
    const v2f* __restrict__ fMu2, const v2f* __restrict__ fVar2,
    float* __restrict__ accMean, float* __restrict__ accW)
{
    // Per-wave private LDS regions (56 KB total per block).
    __shared__ float sX[2][16 * D_IN];     // staged X tile
    __shared__ float sM[2][16 * NUM_RF];   // m   = scale*relu(m1)
    __shared__ float sE[2][16 * NUM_RF];   // ex2 = m*m + v
    __shared__ float sV[2][16 * NUM_RF];   // v   = scale^2 * v1 * gate

    const int wave  = threadIdx.x >> 5;
    const int lane  = threadIdx.x & 31;
    const int lmod  = lane & 15;           // N / row index within half
    const int lhalf = lane >> 4;           // which 16-lane half
    const int strip = blockIdx.x * 2 + wave;
    const int baseRow = strip * 16;        // 16-row strip, exact cover of N

    float* sXw = sX[wave];
    float* sMw = sM[wave];
    float* sEw = sE[wave];
    float* sVw = sV[wave];

    // ---- Stage X tile [16 x 64] into LDS, coalesced float4 ----
    {
        const float4* g4 = reinterpret_cast<const float4*>(X + (size_t)baseRow * D_IN);
        float4* s4 = reinterpret_cast<float4*>(sXw);
#pragma unroll
        for (int i = 0; i < 8; ++i)
            s4[lane + 32 * i] = g4[lane + 32 * i];
    }
    __syncthreads();

    // ---- A-fragments for X and X^2 (ISA 16x4 f32 A layout):
    // lane L: row = L%16, reg j holds K = 4*kk + 2*(L>=16) + j.
    v2f xa[16], xs[16];
#pragma unroll
    for (int kk = 0; kk < 16; ++kk) {
        const float* p = sXw + lmod * D_IN + kk * 4 + lhalf * 2; // 8B aligned
        v2f a = *reinterpret_cast<const v2f*>(p);
        xa[kk] = a;
        xs[kk] = a * a;
    }

    // ---- Layer 1: m1 = X@Wmu1, v1 = X^2@Wvar1  (K=64 -> 16 WMMA steps) ----
    for (int n1 = 0; n1 < 8; ++n1) {
        const int ncol = n1 * 16 + lmod;
        v8f accm = {};
        v8f accv = {};
#pragma unroll
        for (int kk = 0; kk < 16; ++kk) {
            v2f bm, bv; // B 4x16 layout: reg j = row K=krow+j, N = lane%16
            if constexpr (FRAG) {
                bm = fMu1[(n1 * 16 + kk) * 32 + lane];
                bv = fVar1[(n1 * 16 + kk) * 32 + lane];
            } else {
                const int krow = kk * 4 + lhalf * 2;
                bm.x = Wmu1[krow * NUM_RF + ncol];
                bm.y = Wmu1[(krow + 1) * NUM_RF + ncol];
                bv.x = Wvar1[krow * NUM_RF + ncol];
                bv.y = Wvar1[(krow + 1) * NUM_RF + ncol];
            }
            accm = wmma_f32_16x16x4(xa[kk], bm, accm);
            accv = wmma_f32_16x16x4(xs[kk], bv, accv);
        }
        // C/D layout: reg r -> row r + 8*lhalf, col = lane%16.
#pragma unroll
        for (int r = 0; r < 8; ++r) {
            const int row = r + lhalf * 8;
            const float m1 = accm[r];
            const float gate = (m1 > 0.0f) ? 1.0f : 0.0f;
            const float m = RF_SCALE * (m1 > 0.0f ? m1 : 0.0f);
            const float v = RF_SCALE2 * accv[r] * gate;
            const int o = row * NUM_RF + ncol;
            sMw[o] = m;
            sEw[o] = m * m + v;
            sVw[o] = v;
        }
    }
    __syncthreads();

    // Segment ids for this lane's output rows (same for every n-tile).
    int sidx[8];
#pragma unroll
    for (int r = 0; r < 8; ++r)
        sidx[r] = Xidx[baseRow + r + lhalf * 8];

    // ---- Layer 2: m2 = m@Wmu2; v2 = ex2@Wvar2 + v@(Wmu2^2)  (K=128) ----
    for (int n2 = 0; n2 < 4; ++n2) {
        const int ncol = n2 * 16 + lmod;
        v8f am = {};
        v8f ae = {};
        v8f av = {};
#pragma unroll
        for (int kk = 0; kk < 32; ++kk) {
            const int krow = kk * 4 + lhalf * 2;
            const int aoff = lmod * NUM_RF + krow;  // A-frag from LDS planes
            v2f Am = *reinterpret_cast<const v2f*>(sMw + aoff);
            v2f Ae = *reinterpret_cast<const v2f*>(sEw + aoff);
            v2f Av = *reinterpret_cast<const v2f*>(sVw + aoff);
            v2f bm, bv;
            if constexpr (FRAG) {
                bm = fMu2[(n2 * 32 + kk) * 32 + lane];
                bv = fVar2[(n2 * 32 + kk) * 32 + lane];
            } else {
                bm.x = Wmu2[krow * D_OUT + ncol];
                bm.y = Wmu2[(krow + 1) * D_OUT + ncol];
                bv.x = Wvar2[krow * D_OUT + ncol];
                bv.y = Wvar2[(krow + 1) * D_OUT + ncol];
            }
            v2f bmsq = bm * bm;                     // Wmu2^2 in registers
            am = wmma_f32_16x16x4(Am, bm, am);
            ae = wmma_f32_16x16x4(Ae, bv, ae);
            av = wmma_f32_16x16x4(Av, bmsq, av);
        }
        // Precision-weighted segment accumulation (10 adds/slot on average).
#pragma unroll
        for (int r = 0; r < 8; ++r) {
            const float m2 = am[r];
            const float v2 = ae[r] + av[r];
            const float prec = 1.0f / (v2 + EPSF);
            const float w = prec * m2;
            const size_t o = (size_t)sidx[r] * D_OUT + ncol;
            unsafeAtomicAdd(accMean + o, w);   // -> global_atomic_add_f32
            unsafeAtomicAdd(accW + o, prec);
        }
    }
}

// out[0:half)      holds sum(prec*m)  -> becomes mean_sum * var_i
// out[half:2half)  holds sum(prec)    -> becomes var_i = 1/(w_sum + eps)
__global__ __launch_bounds__(256) void dgp_finalize_kernel(float* __restrict__ out,
                                                           int half)
{
    const int i = blockIdx.x * 256 + threadIdx.x;
    if (i < half) {
        const float w = out[half + i];
        const float var_i = 1.0f / (w + EPSF);
        out[i] = out[i] * var_i;
        out[half + i] = var_i;
    }
}

extern "C" void kernel_launch(void* const* d_in, const int* in_sizes, int n_in,
                              void* d_out, int out_size, void* d_ws, size_t ws_size,
                              hipStream_t stream) {
    (void)in_sizes; (void)n_in;
    const float* X     = (const float*)d_in[0];
    const int*   Xidx  = (const int*)d_in[1];
    const float* Wmu1  = (const float*)d_in[2];
    const float* Wvar1 = (const float*)d_in[3];
    const float* Wmu2  = (const float*)d_in[4];
    const float* Wvar2 = (const float*)d_in[5];
    float* out = (float*)d_out;
    const int half = U_SEG * D_OUT;   // == out_size / 2

    // Zero both accumulator halves (graph-capture safe).
    hipMemsetAsync(d_out, 0, (size_t)out_size * sizeof(float), stream);

    const size_t fragBytes = (size_t)(FRAG1_PAIRS * 2 + FRAG2_PAIRS * 2) * sizeof(v2f);
    const bool useFrag = (d_ws != nullptr) && (ws_size >= fragBytes);

    v2f* fMu1  = (v2f*)d_ws;
    v2f* fVar1 = fMu1 + FRAG1_PAIRS;
    v2f* fMu2  = fVar1 + FRAG1_PAIRS;
    v2f* fVar2 = fMu2 + FRAG2_PAIRS;

    if (useFrag) {
        // One-time (per call) weight swizzle: 8192 fragment pairs.
        dgp_fragmentize_weights<<<32, 256, 0, stream>>>(
            Wmu1, Wvar1, Wmu2, Wvar2, fMu1, fVar1, fMu2, fVar2);
        dgp_fused_kernel<true><<<N_ROWS / 32, 64, 0, stream>>>(
            X, Xidx, Wmu1, Wvar1, Wmu2, Wvar2,
            fMu1, fVar1, fMu2, fVar2, out, out + half);
    } else {
        dgp_fused_kernel<false><<<N_ROWS / 32, 64, 0, stream>>>(
            X, Xidx, Wmu1, Wvar1, Wmu2, Wvar2,
            fMu1, fVar1, fMu2, fVar2, out, out + half);
    }

    dgp_finalize_kernel<<<(half + 255) / 256, 256, 0, stream>>>(out, half);
}